// MaskPatchClassificationHead_59863254172618
// MI455X (gfx1250) — compile-verified
//
#include <hip/hip_runtime.h>
#include <hip/hip_bf16.h>
#include <math.h>

// Problem constants (match reference)
#define BB    4
#define PP    256
#define FF    512          // F_IN == F_OUT == 512
#define TAU   0.07f
#define EPS_C 1e-5f

typedef float v2f __attribute__((ext_vector_type(2)));
typedef float v8f __attribute__((ext_vector_type(8)));

// ---------------------------------------------------------------------------
// 1) eq[b,p,q] = 1 if label rows p,q identical (d_L1 == 0), else 0
//    One block per (b,p); row p cached in LDS; thread q scans its row.
// ---------------------------------------------------------------------------
__global__ void eq_kernel(const float* __restrict__ labels, float* __restrict__ EQ) {
    const int bp = blockIdx.x;            // 0..1023
    const int b  = bp >> 8;
    const int p  = bp & 255;
    const int tid = threadIdx.x;          // 0..255

    __shared__ float row[FF];
    const float* lb = labels + (size_t)b * PP * FF;
    row[tid]       = lb[(size_t)p * FF + tid];
    row[tid + 256] = lb[(size_t)p * FF + tid + 256];
    __syncthreads();

    const int q = tid;
    const float4* qr = (const float4*)(lb + (size_t)q * FF);
    int eq = 1;
#pragma unroll 4
    for (int f = 0; f < FF / 4; ++f) {
        float4 v = qr[f];
        eq &= (v.x == row[4 * f + 0]) & (v.y == row[4 * f + 1]) &
              (v.z == row[4 * f + 2]) & (v.w == row[4 * f + 3]);
    }
    EQ[(size_t)bp * PP + q] = eq ? 1.0f : 0.0f;
}

// ---------------------------------------------------------------------------
// 2) h = X (1024x512) * W^T (512x512) + bias, via V_WMMA_F32_16X16X4_F32.
//    One wave -> one 16x16 output tile. K-loop in steps of 4.
//    A frag (16x4 f32): lane L(0..15): M=L, K=k..k+1 ; lane L(16..31): M=L-16, K=k+2..k+3
//    B frag (4x16 f32): v0/v1 hold rows K=koff / koff+1 at N=lane%16, with
//    koff = 2*(lane>=16). B(k,n) = W[n,k] (W is row-major [out,in]), so a
//    float2 load at W + n*512 + k + koff supplies both fragment VGPRs.
// ---------------------------------------------------------------------------
__global__ void gemm_fc(const float* __restrict__ X, const float* __restrict__ W,
                        const float* __restrict__ bias, float* __restrict__ H) {
    const int wave = (blockIdx.x * blockDim.x + threadIdx.x) >> 5;  // 0..2047
    const int lane = threadIdx.x & 31;
    const int tm = (wave >> 5) << 4;      // 64 M-tiles
    const int tn = (wave & 31) << 4;      // 32 N-tiles
    const int ln   = lane & 15;
    const int koff = (lane >> 4) << 1;

    const float* arow = X + (size_t)(tm + ln) * FF + koff;
    const float* brow = W + (size_t)(tn + ln) * FF + koff;

    v8f acc = {};
#pragma unroll 8
    for (int k = 0; k < FF; k += 4) {
        v2f a  = *(const v2f*)(arow + k);
        v2f bb = *(const v2f*)(brow + k);
        acc = __builtin_amdgcn_wmma_f32_16x16x4_f32(
            /*neg_a=*/false, a, /*neg_b=*/false, bb,
            /*c_mod=*/(short)0, acc, /*reuse_a=*/false, /*reuse_b=*/false);
    }

    // C/D layout: VGPR r, lanes 0-15 -> M=r, lanes 16-31 -> M=r+8; N = lane%16
    const float bv   = bias[tn + ln];
    const int mbase  = tm + ((lane >> 4) << 3);
    const int col    = tn + ln;
#pragma unroll
    for (int r = 0; r < 8; ++r)
        H[(size_t)(mbase + r) * FF + col] = acc[r] + bv;
}

// ---------------------------------------------------------------------------
// 3) In-place row L2-normalization of H (1024 rows x 512), torch-style
//    x = h / max(||h||, 1e-12)
// ---------------------------------------------------------------------------
__global__ void norm_rows(float* __restrict__ XH) {
    const int r   = blockIdx.x;           // 0..1023
    const int tid = threadIdx.x;          // 0..255
    __shared__ float red[256];

    float* row = XH + (size_t)r * FF;
    const float v0 = row[tid];
    const float v1 = row[tid + 256];
    red[tid] = v0 * v0 + v1 * v1;
    __syncthreads();
    for (int off = 128; off > 0; off >>= 1) {
        if (tid < off) red[tid] += red[tid + off];
        __syncthreads();
    }
    const float inv = 1.0f / fmaxf(sqrtf(red[0]), 1e-12f);
    row[tid]       = v0 * inv;
    row[tid + 256] = v1 * inv;
}

// ---------------------------------------------------------------------------
// 4) scores[b] = (x_b * x_b^T) / TAU via WMMA f32 16x16x4. 256 tiles/batch.
// ---------------------------------------------------------------------------
__global__ void gemm_scores(const float* __restrict__ Xn, float* __restrict__ SC) {
    const int wave = (blockIdx.x * blockDim.x + threadIdx.x) >> 5;  // 0..1023
    const int lane = threadIdx.x & 31;
    const int b    = wave >> 8;
    const int t    = wave & 255;
    const int tm = (t >> 4) << 4;
    const int tn = (t & 15) << 4;
    const int ln   = lane & 15;
    const int koff = (lane >> 4) << 1;

    const float* Xb   = Xn + (size_t)b * PP * FF;
    const float* arow = Xb + (size_t)(tm + ln) * FF + koff;
    const float* brow = Xb + (size_t)(tn + ln) * FF + koff;

    v8f acc = {};
#pragma unroll 8
    for (int k = 0; k < FF; k += 4) {
        v2f a  = *(const v2f*)(arow + k);
        v2f bb = *(const v2f*)(brow + k);
        acc = __builtin_amdgcn_wmma_f32_16x16x4_f32(
            false, a, false, bb, (short)0, acc, false, false);
    }

    float* out = SC + (size_t)b * PP * PP;
    const int   mbase = tm + ((lane >> 4) << 3);
    const int   col   = tn + ln;
    const float itau  = 1.0f / TAU;
#pragma unroll
    for (int r = 0; r < 8; ++r)
        out[(size_t)(mbase + r) * PP + col] = acc[r] * itau;
}

// ---------------------------------------------------------------------------
// 5) Per-row: softmax over 256 scores, clip, log; dist from eq row; KL term.
//    One block per (b,p); deterministic LDS tree reductions.
// ---------------------------------------------------------------------------
__global__ void row_loss(const float* __restrict__ SC, const float* __restrict__ EQ,
                         float* __restrict__ RL) {
    const int bp  = blockIdx.x;           // 0..1023
    const int tid = threadIdx.x;          // 0..255
    __shared__ float red[256];

    const float s = SC[(size_t)bp * PP + tid];

    // max
    red[tid] = s;
    __syncthreads();
    for (int off = 128; off > 0; off >>= 1) {
        if (tid < off) red[tid] = fmaxf(red[tid], red[tid + off]);
        __syncthreads();
    }
    const float m = red[0];
    __syncthreads();

    // sum of exp
    const float e = expf(s - m);
    red[tid] = e;
    __syncthreads();
    for (int off = 128; off > 0; off >>= 1) {
        if (tid < off) red[tid] += red[tid + off];
        __syncthreads();
    }
    const float Z = red[0];
    __syncthreads();

    float prob = e / Z;
    prob = fminf(fmaxf(prob, EPS_C), 1.0f - EPS_C);
    const float lp = logf(prob);

    // target distribution from equality mask
    const float eqv = EQ[(size_t)bp * PP + tid];
    red[tid] = eqv;
    __syncthreads();
    for (int off = 128; off > 0; off >>= 1) {
        if (tid < off) red[tid] += red[tid + off];
        __syncthreads();
    }
    const float cnt = red[0];             // >= 1 (diagonal always matches)
    __syncthreads();

    float contrib = 0.0f;
    if (eqv > 0.0f) {
        const float d = eqv / cnt;
        contrib = d * (logf(d) - lp);
    }
    red[tid] = contrib;
    __syncthreads();
    for (int off = 128; off > 0; off >>= 1) {
        if (tid < off) red[tid] += red[tid + off];
        __syncthreads();
    }
    if (tid == 0) RL[bp] = red[0];
}

// ---------------------------------------------------------------------------
// 6) Final deterministic reduction of 1024 row losses; batchmean divide.
// ---------------------------------------------------------------------------
__global__ void final_reduce(const float* __restrict__ RL, float* __restrict__ out) {
    const int tid = threadIdx.x;          // 0..255
    __shared__ float red[256];
    red[tid] = RL[tid] + RL[tid + 256] + RL[tid + 512] + RL[tid + 768];
    __syncthreads();
    for (int off = 128; off > 0; off >>= 1) {
        if (tid < off) red[tid] += red[tid + off];
        __syncthreads();
    }
    if (tid == 0) out[0] = red[0] * (1.0f / (float)(BB * PP));
}

// ---------------------------------------------------------------------------
// Launch: inputs order = {inputs, labels, W, b}; output = scalar loss (f32).
// Workspace layout (floats): XH[1024*512] | SC[4*256*256] | EQ[4*256*256] | RL[1024]
// ---------------------------------------------------------------------------
extern "C" void kernel_launch(void* const* d_in, const int* in_sizes, int n_in,
                              void* d_out, int out_size, void* d_ws, size_t ws_size,
                              hipStream_t stream) {
    const float* inputs = (const float*)d_in[0];
    const float* labels = (const float*)d_in[1];
    const float* W      = (const float*)d_in[2];
    const float* bias   = (const float*)d_in[3];
    float* out = (float*)d_out;

    float* ws = (float*)d_ws;
    float* XH = ws;                         // 524288 floats
    float* SC = ws + 524288;                // 262144 floats
    float* EQ = ws + 786432;                // 262144 floats
    float* RL = ws + 1048576;               // 1024 floats

    eq_kernel  <<<BB * PP, 256, 0, stream>>>(labels, EQ);
    gemm_fc    <<<256,     256, 0, stream>>>(inputs, W, bias, XH);   // 2048 waves
    norm_rows  <<<BB * PP, 256, 0, stream>>>(XH);
    gemm_scores<<<128,     256, 0, stream>>>(XH, SC);                // 1024 waves
    row_loss   <<<BB * PP, 256, 0, stream>>>(SC, EQ, RL);
    final_reduce<<<1,      256, 0, stream>>>(RL, out);
}